// Foo_22411139350828
// MI455X (gfx1250) — compile-verified
//
#include <hip/hip_runtime.h>

typedef __attribute__((ext_vector_type(2))) float v2f;
typedef __attribute__((ext_vector_type(8))) float v8f;

#define LN_EPS 1e-5f
#define DD 5
#define RPB 256                 // rows per block
#define FPB (RPB * DD)          // floats per block = 1280

// LayerNorm(D=5) + Linear(5x5, y = h @ W^T + b) + ReLU, fp32 throughout.
// WMMA path: D = A(16x4 f32) x B(4x16 f32) + C, split K=5 into K=4 + K=1.
__launch_bounds__(RPB)
__global__ void lnlin_wmma_kernel(const float* __restrict__ x,
                                  const float* __restrict__ gamma,
                                  const float* __restrict__ beta,
                                  const float* __restrict__ W,
                                  const float* __restrict__ bias,
                                  float* __restrict__ out)
{
    __shared__ float xs[FPB];   // staged x; reused for y after barrier
    __shared__ float hs[FPB];   // normalized rows
    __shared__ float sW[28];
    __shared__ float sG[DD];
    __shared__ float sB[DD];
    __shared__ float sBias[DD];

    const int t = threadIdx.x;
    const long long base = (long long)blockIdx.x * FPB;

    // ---- coalesced global -> LDS stage (5 x b32 per lane, fully coalesced) ----
    #pragma unroll
    for (int j = 0; j < DD; ++j)
        xs[j * RPB + t] = x[base + j * RPB + t];

    if (t < 25)                       sW[t]        = W[t];
    else if (t >= 32 && t < 32 + DD)  sG[t - 32]   = gamma[t - 32];
    else if (t >= 64 && t < 64 + DD)  sB[t - 64]   = beta[t - 64];
    else if (t >= 96 && t < 96 + DD)  sBias[t - 96] = bias[t - 96];
    __syncthreads();

    // ---- LayerNorm, one row per thread (stride-5 LDS reads: conflict-free) ----
    {
        const int rb = t * DD;
        float v0 = xs[rb + 0], v1 = xs[rb + 1], v2 = xs[rb + 2];
        float v3 = xs[rb + 3], v4 = xs[rb + 4];
        float mean = (v0 + v1 + v2 + v3 + v4) * 0.2f;
        float d0 = v0 - mean, d1 = v1 - mean, d2 = v2 - mean;
        float d3 = v3 - mean, d4 = v4 - mean;
        float var = (d0*d0 + d1*d1 + d2*d2 + d3*d3 + d4*d4) * 0.2f;
        float rs = __frsqrt_rn(var + LN_EPS);
        hs[rb + 0] = d0 * rs * sG[0] + sB[0];
        hs[rb + 1] = d1 * rs * sG[1] + sB[1];
        hs[rb + 2] = d2 * rs * sG[2] + sB[2];
        hs[rb + 3] = d3 * rs * sG[3] + sB[3];
        hs[rb + 4] = d4 * rs * sG[4] + sB[4];
    }
    __syncthreads();

    // ---- y = h @ W^T + b via V_WMMA_F32_16X16X4_F32 ----
    const int lane = t & 31;
    const int wave = t >> 5;
    const int half = lane >> 4;   // 0: lanes 0-15, 1: lanes 16-31
    const int n    = lane & 15;   // output column; valid when n < 5
    const bool nv  = (n < DD);
    const int  nc  = nv ? n : 0;

    // B (4x16): VGPR0 = rows K0 (lanes 0-15) / K2 (lanes 16-31); VGPR1 = K1/K3.
    // B[k][n] = W^T[k][n] = W[n*5 + k].
    v2f B1, B2;
    B1.x = nv ? sW[nc * DD + (half ? 2 : 0)] : 0.0f;
    B1.y = nv ? sW[nc * DD + (half ? 3 : 1)] : 0.0f;
    B2.x = (nv && half == 0) ? sW[nc * DD + 4] : 0.0f;  // K=4 column of W^T
    B2.y = 0.0f;
    const float bn = nv ? sBias[nc] : 0.0f;

    #pragma unroll
    for (int tile = 0; tile < 2; ++tile) {
        const int m0   = wave * 32 + tile * 16;  // 16-row tile base within block
        const int arow = m0 + n;                 // A-layout: M = lane&15 both halves
        const int k0   = half * 2;               // VGPR0 holds K0|K2, VGPR1 K1|K3
        v2f A1, A2;
        A1.x = hs[arow * DD + k0];
        A1.y = hs[arow * DD + k0 + 1];
        A2.x = half ? 0.0f : hs[arow * DD + 4];  // K=4 element in (k=0, lanes 0-15)
        A2.y = 0.0f;

        v8f acc = { bn, bn, bn, bn, bn, bn, bn, bn };
        acc = __builtin_amdgcn_wmma_f32_16x16x4_f32(false, A2, false, B2,
                                                    (short)0, acc, false, false);
        acc = __builtin_amdgcn_wmma_f32_16x16x4_f32(false, A1, false, B1,
                                                    (short)0, acc, false, false);

        // D layout: VGPR r -> row m0 + half*8 + r, column n = lane&15.
        if (nv) {
            #pragma unroll
            for (int r = 0; r < 8; ++r) {
                const int m = m0 + half * 8 + r;
                float v = acc[r];
                xs[m * DD + n] = v > 0.0f ? v : 0.0f;   // ReLU
            }
        }
    }
    __syncthreads();

    // ---- coalesced LDS -> global store ----
    #pragma unroll
    for (int j = 0; j < DD; ++j)
        out[base + j * RPB + t] = xs[j * RPB + t];
}

// Scalar tail for row counts not divisible by RPB (not hit for N = 8388608).
__global__ void lnlin_tail_kernel(const float* __restrict__ x,
                                  const float* __restrict__ gamma,
                                  const float* __restrict__ beta,
                                  const float* __restrict__ W,
                                  const float* __restrict__ bias,
                                  float* __restrict__ out,
                                  long long row0, long long nRows)
{
    long long i = row0 + (long long)blockIdx.x * blockDim.x + threadIdx.x;
    if (i >= row0 + nRows) return;
    const float* xr = x + i * DD;
    float v[DD];
    float mean = 0.0f;
    #pragma unroll
    for (int j = 0; j < DD; ++j) { v[j] = xr[j]; mean += v[j]; }
    mean *= 0.2f;
    float var = 0.0f;
    #pragma unroll
    for (int j = 0; j < DD; ++j) { v[j] -= mean; var += v[j] * v[j]; }
    var *= 0.2f;
    float rs = __frsqrt_rn(var + LN_EPS);
    float h[DD];
    #pragma unroll
    for (int j = 0; j < DD; ++j) h[j] = v[j] * rs * gamma[j] + beta[j];
    #pragma unroll
    for (int o = 0; o < DD; ++o) {
        float y = bias[o];
        #pragma unroll
        for (int k = 0; k < DD; ++k) y += h[k] * W[o * DD + k];
        out[i * DD + o] = y > 0.0f ? y : 0.0f;
    }
}

extern "C" void kernel_launch(void* const* d_in, const int* in_sizes, int n_in,
                              void* d_out, int out_size, void* d_ws, size_t ws_size,
                              hipStream_t stream) {
    (void)n_in; (void)out_size; (void)d_ws; (void)ws_size;
    const float* x     = (const float*)d_in[0];
    const float* gamma = (const float*)d_in[1];
    const float* beta  = (const float*)d_in[2];
    const float* W     = (const float*)d_in[3];
    const float* b     = (const float*)d_in[4];
    float* out = (float*)d_out;

    const long long N = (long long)in_sizes[0] / DD;
    const long long nFull = N / RPB;          // full 256-row blocks (WMMA path)
    const long long tail  = N - nFull * RPB;

    if (nFull > 0) {
        lnlin_wmma_kernel<<<(unsigned)nFull, RPB, 0, stream>>>(x, gamma, beta, W, b, out);
    }
    if (tail > 0) {
        const int tb = 256;
        const unsigned tg = (unsigned)((tail + tb - 1) / tb);
        lnlin_tail_kernel<<<tg, tb, 0, stream>>>(x, gamma, beta, W, b, out,
                                                 nFull * RPB, tail);
    }
}